// SparseRNN_58171037057791
// MI455X (gfx1250) — compile-verified
//
#include <hip/hip_runtime.h>
#include <hip/hip_bf16.h>
#include <math.h>

// Problem constants (match reference)
#define B_  64
#define T_  128
#define I_  1024
#define H_  4096
#define NNZ_IH 131072
#define NNZ_HH 262144
#define TB_ (T_ * B_)   // 8192 GEMM "N" columns = (t,b) pairs

typedef __attribute__((ext_vector_type(2))) float v2f;
typedef __attribute__((ext_vector_type(8))) float v8f;

// ---------------- utility kernels ----------------

__global__ void zero_f32(float* __restrict__ p, int n) {
  int i = blockIdx.x * blockDim.x + threadIdx.x;
  int stride = gridDim.x * blockDim.x;
  for (; i < n; i += stride) p[i] = 0.0f;
}

__global__ void hist_rows(const int* __restrict__ rows, int nnz, int* __restrict__ cnt) {
  int i = blockIdx.x * blockDim.x + threadIdx.x;
  int stride = gridDim.x * blockDim.x;
  for (; i < nnz; i += stride) atomicAdd(&cnt[rows[i]], 1);
}

// Exclusive scan over H=4096 counts, 1 block of 1024 threads, 4 elems/thread.
__global__ void scan_rows(const int* __restrict__ cnt, int* __restrict__ rowptr,
                          int* __restrict__ cursor) {
  __shared__ int s[1024];
  int tid = threadIdx.x;
  int base = tid * 4;
  int v0 = cnt[base + 0], v1 = cnt[base + 1], v2 = cnt[base + 2], v3 = cnt[base + 3];
  int tsum = v0 + v1 + v2 + v3;
  s[tid] = tsum;
  __syncthreads();
  for (int off = 1; off < 1024; off <<= 1) {
    int x = (tid >= off) ? s[tid - off] : 0;
    __syncthreads();
    s[tid] += x;
    __syncthreads();
  }
  int e = s[tid] - tsum;  // exclusive prefix of this thread's chunk
  int p0 = e, p1 = e + v0, p2 = e + v0 + v1, p3 = e + v0 + v1 + v2;
  rowptr[base + 0] = p0; rowptr[base + 1] = p1;
  rowptr[base + 2] = p2; rowptr[base + 3] = p3;
  cursor[base + 0] = p0; cursor[base + 1] = p1;
  cursor[base + 2] = p2; cursor[base + 3] = p3;
  if (tid == 1023) rowptr[H_] = s[1023];
}

__global__ void scatter_csr(const int* __restrict__ rows, const int* __restrict__ cols,
                            const float* __restrict__ vals, int nnz,
                            int* __restrict__ cursor, int* __restrict__ colS,
                            float* __restrict__ valS) {
  int i = blockIdx.x * blockDim.x + threadIdx.x;
  int stride = gridDim.x * blockDim.x;
  for (; i < nnz; i += stride) {
    int pos = atomicAdd(&cursor[rows[i]], 1);
    colS[pos] = cols[i];
    valS[pos] = vals[i];
  }
}

__global__ void densify_ih(const int* __restrict__ rows, const int* __restrict__ cols,
                           const float* __restrict__ vals, int nnz,
                           float* __restrict__ dense) {
  int i = blockIdx.x * blockDim.x + threadIdx.x;
  int stride = gridDim.x * blockDim.x;
  for (; i < nnz; i += stride)
    atomicAdd(&dense[(size_t)rows[i] * I_ + cols[i]], vals[i]);
}

// ---------------- WMMA input-projection GEMM ----------------
// preX[t][h][b] = sum_i Wih_dense[h][i] * x[b][t][i]
// D(16x16) tiles over (H x TB), K=I in chunks of 4, fp32 WMMA (exact vs ref).
// One wave per block; EXEC is all-1s (no divergence) as WMMA requires.
__global__ void __launch_bounds__(32) wmma_preX(const float* __restrict__ Wih,
                                                const float* __restrict__ x,
                                                float* __restrict__ preX) {
  const int l  = threadIdx.x;     // 0..31
  const int lm = l & 15;          // lane within half
  const int hi = l >> 4;          // 0: K/M low half, 1: high half
  const int m0 = blockIdx.x * 16; // H tile
  const int n  = blockIdx.y * 16 + lm; // TB column for this lane
  const int t  = n / B_;
  const int bb = n % B_;

  // A 16x4 fp32: lane(0..15)=row M, vgpr0=K(0|2), vgpr1=K(1|3)
  const float* arow = Wih + (size_t)(m0 + lm) * I_ + 2 * hi;
  // B 4x16 fp32: lane(0..15)=col N, vgpr0=K(0|2), vgpr1=K(1|3); x is (B,T,I) row-major
  const float* bcol = x + (size_t)bb * (T_ * I_) + (size_t)t * I_ + 2 * hi;

  v8f acc = {};
  for (int k = 0; k < I_; k += 4) {
    v2f a, b;
    a.x = arow[k];     a.y = arow[k + 1];
    b.x = bcol[k];     b.y = bcol[k + 1];
    // 8 args: (neg_a, A, neg_b, B, c_mod, C, reuse_a, reuse_b)
    acc = __builtin_amdgcn_wmma_f32_16x16x4_f32(false, a, false, b, (short)0, acc,
                                                false, false);
  }

  // D 16x16: vgpr j holds M=j (lanes 0-15) / M=j+8 (lanes 16-31), N = lane&15
  float* o = preX + (size_t)t * (H_ * B_) + (size_t)(m0 + 8 * hi) * B_ + bb;
#pragma unroll
  for (int j = 0; j < 8; j++) o[(size_t)j * B_] = acc[j];
}

// ---------------- recurrent step ----------------
// One wave per output row r; lanes cover batch (2 floats each, coalesced).
__global__ void __launch_bounds__(256) rnn_step(
    int t, const float* __restrict__ hprev, float* __restrict__ hnext,
    const int* __restrict__ rp_hh, const int* __restrict__ col_hh,
    const float* __restrict__ val_hh, const float* __restrict__ bias,
    const float* __restrict__ preX, int usePre,
    const int* __restrict__ rp_ih, const int* __restrict__ col_ih,
    const float* __restrict__ val_ih, const float* __restrict__ x,
    float* __restrict__ out) {
  const int lane = threadIdx.x & 31;
  const int wave = threadIdx.x >> 5;
  const int r  = blockIdx.x * 8 + wave;
  const int b0 = lane * 2;

  float accx = bias[r];
  float accy = accx;

  if (usePre) {
    const float* p = preX + (size_t)t * (H_ * B_) + (size_t)r * B_ + b0;
    accx += p[0];
    accy += p[1];
  } else {
    // fallback: gather ih term per step from CSR
    const float* x0 = x + (size_t)b0 * (T_ * I_) + (size_t)t * I_;
    const float* x1 = x0 + (size_t)(T_ * I_);
    int k0 = rp_ih[r], k1 = rp_ih[r + 1];
    for (int k = k0; k < k1; k++) {
      int c = col_ih[k];
      float v = val_ih[k];
      accx += v * x0[c];
      accy += v * x1[c];
    }
  }

  // recurrent SpMM: coalesced 256B row gathers of h_prev (L2 resident, 1 MB)
  {
    int k0 = rp_hh[r], k1 = rp_hh[r + 1];
    for (int k = k0; k < k1; k++) {
      int c = col_hh[k];
      float v = val_hh[k];
      const float* m = hprev + (size_t)c * B_ + b0;
      accx += v * m[0];
      accy += v * m[1];
    }
  }

  float hx = tanhf(accx), hy = tanhf(accy);
  float* hn = hnext + (size_t)r * B_ + b0;
  hn[0] = hx;
  hn[1] = hy;
  out[(size_t)b0 * (T_ * H_) + (size_t)t * H_ + r]       = hx;
  out[(size_t)(b0 + 1) * (T_ * H_) + (size_t)t * H_ + r] = hy;
}

// ---------------- launcher ----------------
extern "C" void kernel_launch(void* const* d_in, const int* in_sizes, int n_in,
                              void* d_out, int out_size, void* d_ws, size_t ws_size,
                              hipStream_t stream) {
  const float* x        = (const float*)d_in[0];
  const float* ih_vals  = (const float*)d_in[1];
  const float* hh_vals  = (const float*)d_in[2];
  const float* hh_bias  = (const float*)d_in[3];
  const int*   ih_rows  = (const int*)d_in[4];
  const int*   ih_cols  = (const int*)d_in[5];
  const int*   hh_rows  = (const int*)d_in[6];
  const int*   hh_cols  = (const int*)d_in[7];
  float* out = (float*)d_out;
  char*  ws  = (char*)d_ws;

  size_t off = 0;
  auto alloc = [&](size_t bytes) -> size_t {
    size_t o = off;
    off = (off + bytes + 255) & ~(size_t)255;
    return o;
  };
  size_t o_h0     = alloc((size_t)H_ * B_ * 4);
  size_t o_h1     = alloc((size_t)H_ * B_ * 4);
  size_t o_cnt_hh = alloc((size_t)H_ * 4);
  size_t o_rp_hh  = alloc((size_t)(H_ + 1) * 4);
  size_t o_cur_hh = alloc((size_t)H_ * 4);
  size_t o_col_hh = alloc((size_t)NNZ_HH * 4);
  size_t o_val_hh = alloc((size_t)NNZ_HH * 4);
  size_t o_cnt_ih = alloc((size_t)H_ * 4);
  size_t o_rp_ih  = alloc((size_t)(H_ + 1) * 4);
  size_t o_cur_ih = alloc((size_t)H_ * 4);
  size_t o_col_ih = alloc((size_t)NNZ_IH * 4);
  size_t o_val_ih = alloc((size_t)NNZ_IH * 4);
  size_t o_dense  = alloc((size_t)H_ * I_ * 4);          // 16 MB
  size_t o_preX   = alloc((size_t)T_ * H_ * B_ * 4);     // 134 MB
  size_t fullNeed = off;
  int usePre = (ws_size >= fullNeed) ? 1 : 0;

  float* h0      = (float*)(ws + o_h0);
  float* h1      = (float*)(ws + o_h1);
  int*   cnt_hh  = (int*)(ws + o_cnt_hh);
  int*   rp_hh   = (int*)(ws + o_rp_hh);
  int*   cur_hh  = (int*)(ws + o_cur_hh);
  int*   col_hh  = (int*)(ws + o_col_hh);
  float* val_hh  = (float*)(ws + o_val_hh);
  int*   cnt_ih  = (int*)(ws + o_cnt_ih);
  int*   rp_ih   = (int*)(ws + o_rp_ih);
  int*   cur_ih  = (int*)(ws + o_cur_ih);
  int*   col_ih  = (int*)(ws + o_col_ih);
  float* val_ih  = (float*)(ws + o_val_ih);
  float* dense   = (float*)(ws + o_dense);
  float* preX    = (float*)(ws + o_preX);

  // h0 = 0, count arrays = 0
  zero_f32<<<256, 256, 0, stream>>>(h0, H_ * B_);
  zero_f32<<<16, 256, 0, stream>>>((float*)cnt_hh, H_);
  zero_f32<<<16, 256, 0, stream>>>((float*)cnt_ih, H_);

  // build CSR for both sparse matrices (once; amortized over 128 steps)
  hist_rows<<<512, 256, 0, stream>>>(hh_rows, NNZ_HH, cnt_hh);
  hist_rows<<<512, 256, 0, stream>>>(ih_rows, NNZ_IH, cnt_ih);
  scan_rows<<<1, 1024, 0, stream>>>(cnt_hh, rp_hh, cur_hh);
  scan_rows<<<1, 1024, 0, stream>>>(cnt_ih, rp_ih, cur_ih);
  scatter_csr<<<512, 256, 0, stream>>>(hh_rows, hh_cols, hh_vals, NNZ_HH, cur_hh,
                                       col_hh, val_hh);
  scatter_csr<<<512, 256, 0, stream>>>(ih_rows, ih_cols, ih_vals, NNZ_IH, cur_ih,
                                       col_ih, val_ih);

  if (usePre) {
    // densify W_ih (16 MB) and compute all input projections with fp32 WMMA
    zero_f32<<<4096, 256, 0, stream>>>(dense, H_ * I_);
    densify_ih<<<512, 256, 0, stream>>>(ih_rows, ih_cols, ih_vals, NNZ_IH, dense);
    dim3 g(H_ / 16, TB_ / 16);
    wmma_preX<<<g, 32, 0, stream>>>(dense, x, preX);
  }

  // sequential recurrence, double-buffered hidden state
  for (int t = 0; t < T_; t++) {
    const float* hp = (t & 1) ? h1 : h0;
    float*       hn = (t & 1) ? h0 : h1;
    rnn_step<<<H_ / 8, 256, 0, stream>>>(t, hp, hn, rp_hh, col_hh, val_hh, hh_bias,
                                         preX, usePre, rp_ih, col_ih, val_ih, x, out);
  }
}